// RotatE_69312182222862
// MI455X (gfx1250) — compile-verified
//
#include <hip/hip_runtime.h>
#include <math.h>

// RotatE KGE scoring for MI455X (gfx1250, wave32).
// out = [ -score_sp (32x14541) | -score_po (32x14541) | lhs_norm (32x200) | rhs_norm (32x200) ]

#define N_ENT 14541
#define N_REL 237
#define RANK  200
#define BATCH 32
#define EPB   32          // entities per block (8 waves * 4)
#define EPW   4           // entities per wave
#define TBL   (RANK*BATCH)   // 6400 table entries
#define TWO_PI_F 6.28318530717958647692f

// ---- fast f32 sqrt -> v_sqrt_f32 --------------------------------------------
__device__ __forceinline__ float fsqrt(float v) {
#if __has_builtin(__builtin_amdgcn_sqrtf)
  return __builtin_amdgcn_sqrtf(v);
#else
  return sqrtf(v);
#endif
}

// ---- CDNA5 async global->LDS copy (ASYNCcnt path) ----------------------------
__device__ __forceinline__ void async_copy_b128(void* lds_dst, const void* gsrc) {
  // VDST = LDS byte address (low 32 bits of generic LDS pointer), VADDR = 64-bit global addr
  unsigned lds = (unsigned)(unsigned long long)lds_dst;
  asm volatile("global_load_async_to_lds_b128 %0, %1, off"
               :: "v"(lds), "v"(gsrc) : "memory");
}
__device__ __forceinline__ void wait_async0() {
  asm volatile("s_wait_asynccnt 0" ::: "memory");
}

// ---- per-(b,k) table entry ---------------------------------------------------
// po = {cos, sin, rre, rim}; sp = {qre, qim}
__device__ __forceinline__ void table_entry(
    int idx, const int* __restrict__ x, const float* __restrict__ emb,
    const float* __restrict__ rels,
    float4& po_v, float2& sp_v, float& ln, float& rn)
{
  int b = idx & (BATCH - 1);
  int k = idx >> 5;
  int h = x[b*3 + 0];
  int r = x[b*3 + 1];
  int t = x[b*3 + 2];
  float lre = emb[h*(2*RANK) + k];
  float lim = emb[h*(2*RANK) + RANK + k];
  float rre = emb[t*(2*RANK) + k];
  float rim = emb[t*(2*RANK) + RANK + k];
  float s, c;
  sincosf(rels[r*RANK + k] * TWO_PI_F, &s, &c);
  po_v = make_float4(c, s, rre, rim);
  sp_v = make_float2(lre*c - lim*s, lre*s + lim*c);
  ln = fsqrt(lre*lre + lim*lim);
  rn = fsqrt(rre*rre + rim*rim);
}

// ---- kernel 1: tables + norms ------------------------------------------------
__global__ __launch_bounds__(256) void rotate_prep(
    const int* __restrict__ x, const float* __restrict__ emb,
    const float* __restrict__ rels, float* __restrict__ out,
    float4* __restrict__ po_tab, float2* __restrict__ sp_tab, int write_tabs)
{
  int idx = blockIdx.x * 256 + threadIdx.x;
  if (idx >= TBL) return;
  float4 pv; float2 sv; float ln, rn;
  table_entry(idx, x, emb, rels, pv, sv, ln, rn);
  int b = idx & (BATCH - 1);
  int k = idx >> 5;
  float* nbase = out + (size_t)2 * BATCH * N_ENT;
  nbase[b*RANK + k] = ln;                         // lhs_norm
  nbase[BATCH*RANK + b*RANK + k] = rn;            // rhs_norm
  if (write_tabs) {
    po_tab[idx] = pv;
    sp_tab[idx] = sv;
  }
}

// ---- kernel 2: both score matrices ------------------------------------------
// lane = batch index b (wave32 == BATCH). Each wave handles EPW entities.
template<bool FROM_WS>
__global__ __launch_bounds__(256) void rotate_scores(
    const int* __restrict__ x, const float* __restrict__ emb,
    const float* __restrict__ rels,
    const float4* __restrict__ po_tab, const float2* __restrict__ sp_tab,
    float* __restrict__ out_sp, float* __restrict__ out_po)
{
  __shared__ __align__(16) float4 lds_po[TBL];   // 102400 B
  __shared__ __align__(16) float2 lds_sp[TBL];   //  51200 B

  const int tid = threadIdx.x;

  if (FROM_WS) {
    // Stage tables with CDNA5 async DMA (no VGPR round-trip).
    for (int i = tid; i < TBL; i += 256)
      async_copy_b128(&lds_po[i], &po_tab[i]);
    const float4* sp4 = (const float4*)sp_tab;
    float4* lsp4 = (float4*)lds_sp;
    for (int i = tid; i < TBL/2; i += 256)
      async_copy_b128(&lsp4[i], &sp4[i]);
    wait_async0();
  } else {
    // Fallback: recompute tables directly into LDS (no workspace needed).
    for (int i = tid; i < TBL; i += 256) {
      float4 pv; float2 sv; float ln, rn;
      table_entry(i, x, emb, rels, pv, sv, ln, rn);
      lds_po[i] = pv;
      lds_sp[i] = sv;
    }
  }
  __syncthreads();

  const int lane  = tid & 31;          // = batch index b
  const int wv    = tid >> 5;
  const int eBase = blockIdx.x * EPB + wv * EPW;

  float accS[EPW] = {0.f, 0.f, 0.f, 0.f};
  float accP[EPW] = {0.f, 0.f, 0.f, 0.f};

  const float* rows[EPW];
#pragma unroll
  for (int j = 0; j < EPW; ++j) {
    int e = eBase + j;
    if (e >= N_ENT) e = N_ENT - 1;     // clamp; result discarded at store
    rows[j] = emb + (size_t)e * (2*RANK);
  }

#pragma unroll 2
  for (int k = 0; k < RANK; ++k) {
    float2 qv = lds_sp[k*BATCH + lane];   // {qre, qim}   : conflict-free b64
    float4 pv = lds_po[k*BATCH + lane];   // {c,s,rre,rim}: conflict-free b128
#pragma unroll
    for (int j = 0; j < EPW; ++j) {
      float er = rows[j][k];              // wave-uniform -> broadcast load
      float ei = rows[j][RANK + k];
      // score_sp: | (q - e) |
      float dre = qv.x - er;
      float dim = qv.y - ei;
      accS[j] += fsqrt(dre*dre + dim*dim);
      // score_po: | rot(e) - rhs |
      float pre = er*pv.x - ei*pv.y - pv.z;
      float pim = er*pv.y + ei*pv.x - pv.w;
      accP[j] += fsqrt(pre*pre + pim*pim);
    }
  }

#pragma unroll
  for (int j = 0; j < EPW; ++j) {
    int e = eBase + j;
    if (e < N_ENT) {
      out_sp[(size_t)lane*N_ENT + e] = -accS[j];
      out_po[(size_t)lane*N_ENT + e] = -accP[j];
    }
  }
}

// ---- host launcher -----------------------------------------------------------
extern "C" void kernel_launch(void* const* d_in, const int* in_sizes, int n_in,
                              void* d_out, int out_size, void* d_ws, size_t ws_size,
                              hipStream_t stream) {
  const int*   x    = (const int*)d_in[0];
  const float* emb  = (const float*)d_in[1];
  const float* rels = (const float*)d_in[2];
  float* out = (float*)d_out;

  const size_t need = (size_t)TBL*sizeof(float4) + (size_t)TBL*sizeof(float2); // 153600 B
  const bool use_ws = (d_ws != nullptr) && (ws_size >= need);
  float4* po_tab = (float4*)d_ws;
  float2* sp_tab = (float2*)((char*)d_ws + (size_t)TBL*sizeof(float4));

  rotate_prep<<<(TBL + 255)/256, 256, 0, stream>>>(
      x, emb, rels, out, po_tab, sp_tab, use_ws ? 1 : 0);

  float* out_sp = out;
  float* out_po = out + (size_t)BATCH * N_ENT;
  dim3 grid((N_ENT + EPB - 1) / EPB);   // 455 blocks
  if (use_ws) {
    rotate_scores<true><<<grid, 256, 0, stream>>>(
        x, emb, rels, po_tab, sp_tab, out_sp, out_po);
  } else {
    rotate_scores<false><<<grid, 256, 0, stream>>>(
        x, emb, rels, po_tab, sp_tab, out_sp, out_po);
  }
}